// GroupWiseLinear_41721312314338
// MI455X (gfx1250) — compile-verified
//
#include <hip/hip_runtime.h>
#include <hip/hip_bf16.h>

typedef __attribute__((ext_vector_type(2))) float v2f;
typedef __attribute__((ext_vector_type(8))) float v8f;

#define NUM_CLASS 4096
#define HIDDEN    512
#define N_GROUPS  64
#define BATCH     64

// Scatter inverse permutations into workspace: inv[idx[i]] = i.
__global__ void gwl_invperm_kernel(const int* __restrict__ coIdx,
                                   const int* __restrict__ clIdx,
                                   int* __restrict__ invCo,
                                   int* __restrict__ invCl) {
    int i = blockIdx.x * blockDim.x + threadIdx.x;
    if (i < NUM_CLASS) {
        invCo[coIdx[i]] = i;
    } else if (i < 2 * NUM_CLASS) {
        int j = i - NUM_CLASS;
        invCl[clIdx[j]] = j;
    }
}

// One block per (class-tile ct, half p). 4 waves per block, one batch-tile of
// 16 batches each. Each wave runs a 16x16 (batch x class) f32 WMMA GEMM over
// K=HIDDEN in steps of 4, once per distinct group covering the class tile,
// and stores only the columns belonging to that group.
//
// V_WMMA_F32_16X16X4_F32 operand layout (ISA 7.12.2):
//   A (16x4, M=batch): lane 0-15 -> M=lane, holds K=0,1; lane 16-31 -> M=lane-16, K=2,3
//   B (4x16, N=class): symmetric — lane 0-15 -> N=lane, K=0,1; lane 16-31 -> N=lane-16, K=2,3
//   C/D (16x16): vgpr r, lane L -> M = r + 8*(L>=16), N = L&15
__global__ __launch_bounds__(128) void gwl_wmma_kernel(
    const float* __restrict__ x,
    const float* __restrict__ coW, const float* __restrict__ clW,
    const float* __restrict__ cob, const float* __restrict__ clb,
    const int*   __restrict__ coG, const int*   __restrict__ clG,
    const int*   __restrict__ invCo, const int* __restrict__ invCl,
    float* __restrict__ out)
{
    const int lane = threadIdx.x & 31;
    const int wave = threadIdx.x >> 5;      // batch tile 0..3
    const int ct   = blockIdx.x;            // class tile 0..255
    const int p    = blockIdx.y;            // half: 0 = co, 1 = cl

    const float* W    = p ? clW   : coW;
    const float* bias = p ? clb   : cob;
    const int*   G    = p ? clG   : coG;
    const int*   inv  = p ? invCl : invCo;

    const int col   = lane & 15;            // class column / batch row within tile
    const int kbase = (lane >> 4) << 1;     // 0 for lanes 0-15, 2 for lanes 16-31
    const int c     = ct * 16 + col;        // this lane's class column
    const int my_gid = G[c];                // its group (non-decreasing across cols)
    const int b_row  = wave * 16 + col;     // this lane's batch row for the A tile

    const float* wptr = W + (size_t)c * HIDDEN + kbase;   // B operand: W^T column c
    const float  myBias = bias[c];
    const int    outCol = inv[c];
    float* outBase = out + (size_t)p * (BATCH * NUM_CLASS);

    int j0 = 0;
    while (j0 < 16) {
        // Wave-uniform: group of column j0.
        const int g = __shfl(my_gid, j0, 32);

        // A operand: x[b_row, p*64 + g, :]
        const float* xptr = x + ((size_t)b_row * (2 * N_GROUPS) +
                                 (size_t)p * N_GROUPS + g) * HIDDEN + kbase;

        v8f acc = {};
        #pragma unroll 8
        for (int h = 0; h < HIDDEN; h += 4) {
            v2f a = *(const v2f*)(xptr + h);
            v2f b = *(const v2f*)(wptr + h);
            acc = __builtin_amdgcn_wmma_f32_16x16x4_f32(
                /*neg_a=*/false, a, /*neg_b=*/false, b,
                /*c_mod=*/(short)0, acc, /*reuse_a=*/false, /*reuse_b=*/false);
        }

        // Store only columns that belong to group g (each column written once).
        if (my_gid == g) {
            const int mrow0 = (lane >> 4) * 8;  // 0 or 8
            #pragma unroll
            for (int r = 0; r < 8; ++r) {
                const int b = wave * 16 + mrow0 + r;
                outBase[(size_t)b * NUM_CLASS + outCol] = acc[r] + myBias;
            }
        }

        // Advance past the contiguous run of columns with group g (uniform).
        unsigned long long bal = __ballot(my_gid == g);
        j0 += __popc((unsigned)(bal & 0xFFFFu));
    }
}

extern "C" void kernel_launch(void* const* d_in, const int* in_sizes, int n_in,
                              void* d_out, int out_size, void* d_ws, size_t ws_size,
                              hipStream_t stream) {
    const float* x   = (const float*)d_in[0];
    const float* coW = (const float*)d_in[1];
    const float* clW = (const float*)d_in[2];
    const float* cob = (const float*)d_in[3];
    const float* clb = (const float*)d_in[4];
    const int*   coG = (const int*)d_in[5];
    const int*   clG = (const int*)d_in[6];
    const int*   coI = (const int*)d_in[7];
    const int*   clI = (const int*)d_in[8];
    // d_in[9] = group_len scalar (64), hardcoded.

    int* invCo = (int*)d_ws;
    int* invCl = invCo + NUM_CLASS;

    gwl_invperm_kernel<<<(2 * NUM_CLASS + 255) / 256, 256, 0, stream>>>(
        coI, clI, invCo, invCl);

    dim3 grid(NUM_CLASS / 16, 2);
    gwl_wmma_kernel<<<grid, 128, 0, stream>>>(
        x, coW, clW, cob, clb, coG, clG, invCo, invCl, (float*)d_out);
}